// GRU_81612968558593
// MI455X (gfx1250) — compile-verified
//
#include <hip/hip_runtime.h>
#include <math.h>
#include <stdint.h>

// Problem constants (match reference)
#define BV 32
#define TV 512
#define IV 512
#define HV 512
#define G3 1536           // 3*H
#define SEQ_ELEMS ((size_t)BV * TV * 2 * HV)   // 16,777,216 (hidden section follows)

typedef __attribute__((ext_vector_type(2))) float v2f;
typedef __attribute__((ext_vector_type(8))) float v8f;
typedef __attribute__((ext_vector_type(4))) unsigned int v4u;
typedef __attribute__((ext_vector_type(8))) int v8i;
typedef __attribute__((ext_vector_type(4))) int v4i;

// ---------------------------------------------------------------------------
// Phase 0: zero the h ping-pong buffers and the per-step barrier counters.
// Must run every launch (graph replay reuses the same workspace).
// ---------------------------------------------------------------------------
__global__ void init_ws(float* __restrict__ hT, int* __restrict__ ctr) {
    int i = blockIdx.x * 256 + threadIdx.x;
    if (i < 2 * 2 * TV * BV) hT[i] = 0.0f;      // [dir][buf][k=512][b=32]
    if (i < 2 * TV)          ctr[i] = 0;        // [dir][step]
}

// ---------------------------------------------------------------------------
// Phase 1: gi[z] = X(16384x512) * W_ih[z]^T(512x1536), z in {fwd,bwd}; f32 WMMA.
// Workgroup tile 64(M) x 64(N); 8 waves, each wave owns one 16-row M-subtile
// and two 16-col N-subtiles. K staged in LDS in 16-chunks, transposed so
// A/B fragment reads are bank-friendly.
// A frag (16x4 f32): lanes0-15 v0/v1 = K0/K1, lanes16-31 = K2/K3 (ISA 7.12.2).
// B frag (4x16 f32): mirrored (lanes0-15 rows K0/K1; lanes16-31 rows K2/K3).
// ---------------------------------------------------------------------------
__global__ __launch_bounds__(256)
void gi_gemm(const float* __restrict__ X,
             const float* __restrict__ Wf, const float* __restrict__ Wb,
             float* __restrict__ gi_f, float* __restrict__ gi_b)
{
    const float* W  = (blockIdx.z == 0) ? Wf   : Wb;
    float*       gi = (blockIdx.z == 0) ? gi_f : gi_b;

    __shared__ float Xs[16][66];   // Xs[k][m], padded pitch
    __shared__ float Ws[16][66];   // Ws[k][n] = W[n][k] (B operand, transposed)

    const int tid  = threadIdx.x;
    const int lane = tid & 31;
    const int wave = tid >> 5;
    const int half = lane >> 4;    // 0: lanes0-15, 1: lanes16-31
    const int l16  = lane & 15;

    const int row0 = blockIdx.x * 64;   // over M = B*T
    const int col0 = blockIdx.y * 64;   // over N = 3H
    const int mt    = wave & 3;         // M subtile 0..3
    const int npair = wave >> 2;        // 0/1 -> N cols [npair*32, +32)

    const int lm = tid >> 2;            // 0..63  (tile row to stage)
    const int kg = (tid & 3) * 4;       // 0,4,8,12

    v8f acc0 = {};
    v8f acc1 = {};

    for (int kc = 0; kc < IV; kc += 16) {
        float4 xv = *(const float4*)(X + (size_t)(row0 + lm) * IV + kc + kg);
        float4 wv = *(const float4*)(W + (size_t)(col0 + lm) * IV + kc + kg);
        __syncthreads();                 // previous chunk fully consumed
        Xs[kg + 0][lm] = xv.x; Xs[kg + 1][lm] = xv.y;
        Xs[kg + 2][lm] = xv.z; Xs[kg + 3][lm] = xv.w;
        Ws[kg + 0][lm] = wv.x; Ws[kg + 1][lm] = wv.y;
        Ws[kg + 2][lm] = wv.z; Ws[kg + 3][lm] = wv.w;
        __syncthreads();

        #pragma unroll
        for (int k4 = 0; k4 < 4; ++k4) {
            const int kk = k4 * 4 + half * 2;
            v2f a;  a.x  = Xs[kk][mt * 16 + l16];
                    a.y  = Xs[kk + 1][mt * 16 + l16];
            v2f b0; b0.x = Ws[kk][npair * 32 + l16];
                    b0.y = Ws[kk + 1][npair * 32 + l16];
            v2f b1; b1.x = Ws[kk][npair * 32 + 16 + l16];
                    b1.y = Ws[kk + 1][npair * 32 + 16 + l16];
            acc0 = __builtin_amdgcn_wmma_f32_16x16x4_f32(
                       false, a, false, b0, (short)0, acc0, false, false);
            acc1 = __builtin_amdgcn_wmma_f32_16x16x4_f32(
                       false, a, false, b1, (short)0, acc1, false, false);
        }
    }

    // C/D layout: vgpr r -> (M=r + 8*half, N=l16)
    #pragma unroll
    for (int r = 0; r < 8; ++r) {
        const int m = mt * 16 + r + half * 8;
        const size_t base = (size_t)(row0 + m) * G3 + col0 + npair * 32;
        gi[base + l16]      = acc0[r];
        gi[base + 16 + l16] = acc1[r];
    }
}

// ---------------------------------------------------------------------------
// Phase 2: the sequential scan. 32 persistent workgroups:
//   blockIdx.x>>4 = dir (0 fwd, 1 bwd), blockIdx.x&15 = H-slice of 32 cols.
// Each WG keeps its 64 W_hh rows (i-gate + n-gate for its 32 h-cols) in LDS
// for all 512 steps, staged once by the Tensor Data Mover (2 DMA descriptors,
// TDM LDS-padding gives row pitch 516 dwords -> conflict-free B-frag reads:
// banks (4n + k) mod 64 cover all 64 banks across the two half-waves).
// h lives in L2 as hT[dir][buf][k=512][b=32] (transposed so WMMA A-fragment
// loads are coalesced), ping-ponged per step, with a 16-WG release/acquire
// barrier per direction per step.
// ---------------------------------------------------------------------------
#define WP 516   // padded LDS row pitch (dwords): TDM pads 2 dwords every 256

__global__ __launch_bounds__(256)
void gru_scan(const float* __restrict__ gi_f, const float* __restrict__ gi_b,
              const float* __restrict__ Whh_f, const float* __restrict__ Whh_b,
              const int* __restrict__ lengths,
              float* __restrict__ hT, int* __restrict__ ctr,
              float* __restrict__ out)
{
    const int dir   = blockIdx.x >> 4;
    const int slice = blockIdx.x & 15;
    const int c0    = slice * 32;

    const float* gi  = dir ? gi_b  : gi_f;
    const float* Whh = dir ? Whh_b : Whh_f;
    float* h   = hT + (size_t)dir * 2 * TV * BV;
    int*   bar = ctr + dir * TV;

    // Wlds row n (0..63): n<32 -> W_hh[H + c0 + n][.] (i-gate),
    //                     n>=32 -> W_hh[2H + c0 + n-32][.] (n-gate)
    __shared__ float Wlds[64 * WP];   // 132,096 B
    __shared__ float ghs[32][68];     // gh[b][n] per step

    const int tid  = threadIdx.x;
    const int lane = tid & 31;
    const int wave = tid >> 5;
    const int half = lane >> 4;
    const int l16  = lane & 15;

    // ---- Stage the 64x512 f32 weight slice once via the Tensor Data Mover.
    if (wave == 0) {
        const unsigned lds_base   = (unsigned)(uintptr_t)&Wlds[0];
        const unsigned half_bytes = 32u * WP * 4u;
        #pragma unroll
        for (int blk = 0; blk < 2; ++blk) {
            const int grow0 = (blk == 0) ? (HV + c0) : (2 * HV + c0);
            const uint64_t ga = (uint64_t)(uintptr_t)(Whh + (size_t)grow0 * HV);
            v4u g0;
            g0[0] = 1u;                                  // count=1, user mode
            g0[1] = lds_base + (unsigned)blk * half_bytes;    // D#.lds_addr
            g0[2] = (unsigned)(ga & 0xFFFFFFFFu);        // global_addr[31:0]
            g0[3] = (unsigned)(ga >> 32) | (2u << 30);   // global_addr[56:32] | type=2
            v8i g1;
            g1[0] = 0x00020000 |        // data_size = 2 (4 bytes)
                    (1 << 20)  |        // pad_enable
                    (7 << 22)  |        // pad_interval: every 256 dwords
                    (1 << 25);          // pad_amount: 2 dwords
            g1[1] = (int)(512u << 16);  // tensor_dim0 = 512 (low 16 bits)
            g1[2] = (int)(32u << 16);   // tensor_dim0 hi=0 | tensor_dim1 = 32
            g1[3] = (int)(512u << 16);  // tensor_dim1 hi=0 | tile_dim0 = 512
            g1[4] = 32;                 // tile_dim1 = 32, tile_dim2 = 0
            g1[5] = 512;                // tensor_dim0_stride = 512
            g1[6] = 0; g1[7] = 0;
            v4i g2; g2[0] = 1; g2[1] = 0; g2[2] = 0; g2[3] = 0;  // tensor_dim2=1
            v4i g3; g3[0] = 0; g3[1] = 0; g3[2] = 0; g3[3] = 0;
            v8i g4; g4[0] = 0; g4[1] = 0; g4[2] = 0; g4[3] = 0;  // clang-23 6-arg
            g4[4] = 0; g4[5] = 0; g4[6] = 0; g4[7] = 0;          // form: unused/0
            __builtin_amdgcn_tensor_load_to_lds(g0, g1, g2, g3, g4, 0);
        }
        __builtin_amdgcn_s_wait_tensorcnt(0);
    }
    __syncthreads();

    const int mt = wave & 1;     // batch subtile (16 rows)
    const int nt = wave >> 1;    // gate subtile (16 of 64 cols)
    const int nrow = (nt * 16 + l16) * WP;   // LDS row base for B fragments

    for (int it = 0; it < TV; ++it) {
        const int s = dir ? (TV - 1 - it) : it;          // scan index
        const float* hcur = h + (size_t)(it & 1) * TV * BV;
        float*       hnxt = h + (size_t)((it + 1) & 1) * TV * BV;

        if (it > 0) {
            if (tid == 0) {
                while (__hip_atomic_load(&bar[it - 1], __ATOMIC_ACQUIRE,
                                         __HIP_MEMORY_SCOPE_AGENT) < 16) {
                    __builtin_amdgcn_s_sleep(1);
                }
            }
            __builtin_amdgcn_fence(__ATOMIC_ACQUIRE, "agent");  // drop stale h lines
            __syncthreads();
        }

        // gh = h(32x512) * Wslice^T -> (32x64); one 16x16 tile per wave
        v8f acc = {};
        for (int k4 = 0; k4 < 128; ++k4) {
            const int kk   = k4 * 4 + half * 2;
            const int koff = kk + ((kk >> 8) << 1);   // skip TDM pad dwords
            v2f a, b;
            a.x = hcur[(size_t)kk * BV + mt * 16 + l16];
            a.y = hcur[(size_t)(kk + 1) * BV + mt * 16 + l16];
            b.x = Wlds[nrow + koff];
            b.y = Wlds[nrow + koff + 1];
            acc = __builtin_amdgcn_wmma_f32_16x16x4_f32(
                      false, a, false, b, (short)0, acc, false, false);
        }
        #pragma unroll
        for (int r = 0; r < 8; ++r)
            ghs[mt * 16 + r + half * 8][nt * 16 + l16] = acc[r];
        __syncthreads();

        // Elementwise update for this slice: 32 batches x 32 cols
        #pragma unroll
        for (int j = 0; j < 4; ++j) {
            const int e = tid + j * 256;
            const int b = e >> 5;
            const int c = e & 31;
            const int len = lengths[b];

            const int tsrc  = dir ? ((s + len) & (TV - 1)) : s;
            const float msk = dir ? ((((s + len) & (TV - 1)) < len) ? 1.0f : 0.0f)
                                  : ((s < len) ? 1.0f : 0.0f);

            const float* grow = gi + ((size_t)b * TV + tsrc) * G3;
            const float i_r = grow[c0 + c];
            const float i_i = grow[HV + c0 + c];
            const float i_n = grow[2 * HV + c0 + c];
            const float h_i = ghs[b][c];
            const float h_n = ghs[b][32 + c];
            const float hold = hcur[(size_t)(c0 + c) * BV + b];

            const float g_r = 1.0f / (1.0f + __expf(-i_r));
            const float g_z = 1.0f / (1.0f + __expf(-i_i));
            const float f_z = g_z * (1.0f - g_z);
            const float g_n = tanhf(i_n);
            const float f_n = 1.0f - g_n * g_n;

            float hnew = f_n * (1.0f - g_z) * g_r * h_n
                       - f_z * g_n * h_i
                       + g_z * hold;
            if (it == 0) hnew += (1.0f - g_z) * g_n;   // init_flag term (h==0 here)

            const float hout = msk * hnew + (1.0f - msk) * hold;
            hnxt[(size_t)(c0 + c) * BV + b] = hout;

            int tdst;
            if (dir == 0) tdst = s;
            else          tdst = (TV - 1) - ((TV - 1 - s - len + 2 * TV) & (TV - 1));

            out[((size_t)b * TV + tdst) * (2 * HV) + dir * HV + c0 + c] = hout;
            if (it == TV - 1)
                out[SEQ_ELEMS + (size_t)b * (2 * HV) + dir * HV + c0 + c] = hout;

            // Prefetch next timestep's gi rows into L2/L0 while we wait at the
            // inter-WG barrier (global_prefetch_b8).
            if (it + 1 < TV) {
                const int sn  = dir ? (s - 1) : (s + 1);
                const int tsn = dir ? ((sn + len) & (TV - 1)) : sn;
                const float* gnx = gi + ((size_t)b * TV + tsn) * G3;
                __builtin_prefetch(gnx + c0 + c, 0, 1);
                __builtin_prefetch(gnx + HV + c0 + c, 0, 1);
                __builtin_prefetch(gnx + 2 * HV + c0 + c, 0, 1);
            }
        }

        __builtin_amdgcn_fence(__ATOMIC_RELEASE, "agent");
        __syncthreads();
        if (tid == 0)
            __hip_atomic_fetch_add(&bar[it], 1, __ATOMIC_RELEASE,
                                   __HIP_MEMORY_SCOPE_AGENT);
    }
}

// ---------------------------------------------------------------------------
extern "C" void kernel_launch(void* const* d_in, const int* in_sizes, int n_in,
                              void* d_out, int out_size, void* d_ws, size_t ws_size,
                              hipStream_t stream) {
    (void)in_sizes; (void)n_in; (void)out_size; (void)ws_size;

    const float* x    = (const float*)d_in[0];
    const int*   len  = (const int*)  d_in[1];
    const float* Wihf = (const float*)d_in[2];
    const float* Whhf = (const float*)d_in[3];
    const float* Wihb = (const float*)d_in[4];
    const float* Whhb = (const float*)d_in[5];
    float* out = (float*)d_out;

    // Workspace layout (needs ~202 MB): gi_f | gi_b | h ping-pong | barriers
    float* gi_f = (float*)d_ws;
    float* gi_b = gi_f + (size_t)BV * TV * G3;
    float* hT   = gi_b + (size_t)BV * TV * G3;
    int*   ctr  = (int*)(hT + (size_t)2 * 2 * TV * BV);

    init_ws<<<256, 256, 0, stream>>>(hT, ctr);
    gi_gemm<<<dim3((BV * TV) / 64, G3 / 64, 2), 256, 0, stream>>>(
        x, Wihf, Wihb, gi_f, gi_b);
    gru_scan<<<32, 256, 0, stream>>>(gi_f, gi_b, Whhf, Whhb, len, hT, ctr, out);
}